// Model_50714973831178
// MI455X (gfx1250) — compile-verified
//
#include <hip/hip_runtime.h>

// ============================================================================
// 2-layer GraphSAGE (mean) + edge dot scoring for MI455X (gfx1250, wave32).
//
//   k_zero   : float4-clear deg[N], acc1[4N], acc2[8N]          (52 MB)
//   k_edge1  : acc1[dst] += x[src] (float4 gather + 4 f32 L2 atomics), deg++
//   k_node1  : per wave, 16 nodes:  h1 = x@W1s.T + (acc1/deg)@W1n.T + b1
//              via 2 chained V_WMMA_F32_16X16X4_F32; D tile staged in LDS for
//              (a) coalesced float4 h1 stores and (b) A-layout reuse in the
//              fused projection p2 = h1 @ W2n.T (4 chained WMMAs).
//   k_edge2  : acc2[dst] += p2[src] (2x float4 gather + 8 f32 atomics) --
//              aggregating in the projected 8-dim space halves edge traffic.
//   k_node2  : h2 = h1@W2s.T + acc2/deg + b2 (4 chained WMMAs; mean term in
//              C-init; 1 divide per node shared via LDS; coalesced IO).
//   k_score  : out[e] = dot8(h2[src], h2[dst]), non-temporal stream-out.
//
// Workspace (floats): [deg N | acc1 4N | acc2 8N | h1 16N | p2 8N | h2 8N]
// = 45N floats = 180 MB for N=1M.  All per-phase working sets are L2-resident
// (192 MB), so the edge phases run at L2 bandwidth / atomic throughput.
// ============================================================================

typedef float v2f __attribute__((ext_vector_type(2)));
typedef float v8f __attribute__((ext_vector_type(8)));

#define TPB 256
#define WPB 8   // waves per block (wave32)
#define PT 20   // LDS row pad for 16-wide transpose tile (16B-aligned rows,
                // stride 20 mod 64 banks -> conflict-free column reads)
#define P8 12   // LDS row pad for 8-wide tiles (16B-aligned rows)

// ---------------------------------------------------------------------------
__global__ __launch_bounds__(TPB) void k_zero(float4* __restrict__ p, int n4,
                                              float* __restrict__ tail,
                                              int ntail) {
  int i = blockIdx.x * TPB + threadIdx.x;
  int stride = gridDim.x * TPB;
  float4 z = make_float4(0.f, 0.f, 0.f, 0.f);
  for (; i < n4; i += stride) p[i] = z;
  if (blockIdx.x == 0 && threadIdx.x < ntail) tail[threadIdx.x] = 0.f;
}

// ---------------------------------------------------------------------------
__global__ __launch_bounds__(TPB) void k_edge1(const float4* __restrict__ x,
                                               const int* __restrict__ src,
                                               const int* __restrict__ dst,
                                               float* __restrict__ acc1,
                                               float* __restrict__ deg, int E) {
  int e = blockIdx.x * TPB + threadIdx.x;
  if (e >= E) return;
  int s = src[e], d = dst[e];
  float4 xs = x[s];                   // 16B gather, L2-resident (16 MB)
  atomicAdd(&acc1[d * 4 + 0], xs.x);  // global_atomic_add_f32, L2-resident
  atomicAdd(&acc1[d * 4 + 1], xs.y);
  atomicAdd(&acc1[d * 4 + 2], xs.z);
  atomicAdd(&acc1[d * 4 + 3], xs.w);
  atomicAdd(&deg[d], 1.0f);
}

// ---------------------------------------------------------------------------
// WMMA operand layouts (ISA 7.12.2), 32-bit, wave32:
//   A 16x4 : lanes 0-15 row m=lane {VGPR0=K0,VGPR1=K1}; lanes 16-31 {K2,K3}
//   B 4x16 : col n=lane&15; VGPR0 rows {K0 lo-half, K2 hi-half}, VGPR1 {K1,K3}
//   C/D    : VGPR j = row M=j (lanes 0-15) / M=j+8 (lanes 16-31), N=lane&15
__global__ __launch_bounds__(TPB) void k_node1(
    const float* __restrict__ x, const float* __restrict__ acc1,
    const float* __restrict__ deg, const float* __restrict__ W1s,
    const float* __restrict__ W1n, const float* __restrict__ b1,
    const float* __restrict__ W2n, float* __restrict__ h1,
    float* __restrict__ p2, int numTiles, int N) {
  __shared__ float ldsT[WPB][16][PT];  // h1 tile (transpose bounce)
  __shared__ float ldsP[WPB][16][P8];  // p2 tile (store staging)

  const int lane = threadIdx.x & 31;
  const int w = threadIdx.x >> 5;
  const int tile = blockIdx.x * WPB + w;
  if (tile >= numTiles) return;        // wave-uniform: EXEC all-1s below
  const int lo = lane & 15;
  const int hi = lane >> 4;            // 0 or 1
  const int ks = 2 * hi;               // K offset of this half inside a chunk
  const int base = tile * 16;
  const bool full = (base + 16 <= N);  // wave-uniform fast path

  // ---- layer-1 GEMM: h1 = x@W1s.T + mean@W1n.T + b1 ----
  int m = full ? (base + lo) : min(base + lo, N - 1);
  v2f a_self, a_mean, b_self, b_neigh;
  a_self.x = x[m * 4 + ks];
  a_self.y = x[m * 4 + ks + 1];
  float inv = 1.0f / fmaxf(deg[m], 1.0f);  // one IEEE divide per lane
  a_mean.x = acc1[m * 4 + ks] * inv;
  a_mean.y = acc1[m * 4 + ks + 1] * inv;
  b_self.x = W1s[lo * 4 + ks];             // B[k][n] = W[n][k], W is [16,4]
  b_self.y = W1s[lo * 4 + ks + 1];
  b_neigh.x = W1n[lo * 4 + ks];
  b_neigh.y = W1n[lo * 4 + ks + 1];
  float bb = b1[lo];
  v8f c;
#pragma unroll
  for (int j = 0; j < 8; ++j) c[j] = bb;

  c = __builtin_amdgcn_wmma_f32_16x16x4_f32(false, a_self, false, b_self,
                                            (short)0, c, false, false);
  c = __builtin_amdgcn_wmma_f32_16x16x4_f32(false, a_mean, false, b_neigh,
                                            (short)0, c, false, false);

  // ---- stage D tile (wave-private LDS; same-wave DS ops are in-order) ----
#pragma unroll
  for (int j = 0; j < 8; ++j) ldsT[w][j + 8 * hi][lo] = c[j];
  __builtin_amdgcn_wave_barrier();

  // ---- h1 store: 2 coalesced float4 per lane on the fast path ----
  if (full) {
#pragma unroll
    for (int t = 0; t < 2; ++t) {
      int idx = (lane + 32 * t) * 4;  // 0..252 over the 16x16 tile
      int r = idx >> 4, cc = idx & 15;
      float4 v = *reinterpret_cast<const float4*>(&ldsT[w][r][cc]);
      *reinterpret_cast<float4*>(&h1[(base + r) * 16 + cc]) = v;
    }
  } else {
#pragma unroll
    for (int j = 0; j < 8; ++j) {
      int node = base + j + 8 * hi;
      if (node < N) h1[node * 16 + lo] = c[j];
    }
  }

  // ---- fused projection: p2 = h1_tile @ W2n.T (K=16, cols padded to 16) ----
  v8f dacc = {};
#pragma unroll
  for (int ch = 0; ch < 4; ++ch) {
    v2f a, b;
    a.x = ldsT[w][lo][4 * ch + ks];  // conflict-free: stride PT=20 mod 64
    a.y = ldsT[w][lo][4 * ch + ks + 1];
    b.x = (lo < 8) ? W2n[lo * 16 + 4 * ch + ks] : 0.0f;  // W2n is [8,16]
    b.y = (lo < 8) ? W2n[lo * 16 + 4 * ch + ks + 1] : 0.0f;
    dacc = __builtin_amdgcn_wmma_f32_16x16x4_f32(false, a, false, b, (short)0,
                                                 dacc, false, false);
  }
  __builtin_amdgcn_wave_barrier();
  if (lo < 8) {
#pragma unroll
    for (int j = 0; j < 8; ++j) ldsP[w][j + 8 * hi][lo] = dacc[j];
  }
  __builtin_amdgcn_wave_barrier();
  if (full) {
    int idx = lane * 4;  // 0..124 over the 16x8 tile
    int r = idx >> 3, cc = idx & 7;
    float4 v = make_float4(ldsP[w][r][cc], ldsP[w][r][cc + 1],
                           ldsP[w][r][cc + 2], ldsP[w][r][cc + 3]);
    *reinterpret_cast<float4*>(&p2[base * 8 + idx]) = v;  // fully coalesced
  } else if (lo < 8) {
#pragma unroll
    for (int j = 0; j < 8; ++j) {
      int node = base + j + 8 * hi;
      if (node < N) p2[node * 8 + lo] = dacc[j];
    }
  }
}

// ---------------------------------------------------------------------------
__global__ __launch_bounds__(TPB) void k_edge2(const float4* __restrict__ p2,
                                               const int* __restrict__ src,
                                               const int* __restrict__ dst,
                                               float* __restrict__ acc2, int E) {
  int e = blockIdx.x * TPB + threadIdx.x;
  if (e >= E) return;
  int s = src[e], d = dst[e];
  float4 a = p2[s * 2 + 0];  // 32B gather (projected 8-dim space)
  float4 b = p2[s * 2 + 1];
  atomicAdd(&acc2[d * 8 + 0], a.x);
  atomicAdd(&acc2[d * 8 + 1], a.y);
  atomicAdd(&acc2[d * 8 + 2], a.z);
  atomicAdd(&acc2[d * 8 + 3], a.w);
  atomicAdd(&acc2[d * 8 + 4], b.x);
  atomicAdd(&acc2[d * 8 + 5], b.y);
  atomicAdd(&acc2[d * 8 + 6], b.z);
  atomicAdd(&acc2[d * 8 + 7], b.w);
}

// ---------------------------------------------------------------------------
// h2 = h1 @ W2s.T + acc2/max(deg,1) + b2  (mean term folded into C init)
__global__ __launch_bounds__(TPB) void k_node2(
    const float* __restrict__ h1, const float* __restrict__ acc2,
    const float* __restrict__ deg, const float* __restrict__ W2s,
    const float* __restrict__ b2, float* __restrict__ h2, int numTiles, int N) {
  __shared__ float ldsA[WPB][16][P8];  // acc2-in / h2-out tile staging
  __shared__ float ldsI[WPB][16];      // shared reciprocal degree per row

  const int lane = threadIdx.x & 31;
  const int w = threadIdx.x >> 5;
  const int tile = blockIdx.x * WPB + w;
  if (tile >= numTiles) return;  // wave-uniform: EXEC all-1s below
  const int lo = lane & 15;
  const int hi = lane >> 4;
  const int ks = 2 * hi;
  const int base = tile * 16;
  const bool full = (base + 16 <= N);

  // one divide per node, computed by lanes 0-15 and shared via LDS
  if (lane < 16) {
    int node = full ? (base + lane) : min(base + lane, N - 1);
    ldsI[w][lane] = 1.0f / fmaxf(deg[node], 1.0f);
  }
  // cooperative acc2 tile load: one float4 per lane
  if (full) {
    int idx = lane * 4;
    float4 t = *reinterpret_cast<const float4*>(&acc2[base * 8 + idx]);
    int r = idx >> 3, cc = idx & 7;
    ldsA[w][r][cc + 0] = t.x;
    ldsA[w][r][cc + 1] = t.y;
    ldsA[w][r][cc + 2] = t.z;
    ldsA[w][r][cc + 3] = t.w;
  } else {
#pragma unroll
    for (int q = 0; q < 4; ++q) {
      int ii = lane * 4 + q;
      int r = ii >> 3, cc = ii & 7;
      int node = base + r;
      ldsA[w][r][cc] = (node < N) ? acc2[node * 8 + cc] : 0.0f;
    }
  }
  __builtin_amdgcn_wave_barrier();

  // C init: neighbor mean (pre-projected) + bias; pad cols 8..15 with 0
  float bb = (lo < 8) ? b2[lo] : 0.0f;
  v8f c;
#pragma unroll
  for (int j = 0; j < 8; ++j) {
    int r = j + 8 * hi;
    c[j] = (lo < 8) ? ldsA[w][r][lo] * ldsI[w][r] + bb : 0.0f;
  }

  // K=16 GEMM: 4 chained WMMAs, A from global h1 (L2-hot), B = W2s.T padded
  int m = full ? (base + lo) : min(base + lo, N - 1);
#pragma unroll
  for (int ch = 0; ch < 4; ++ch) {
    v2f a, b;
    a.x = h1[m * 16 + 4 * ch + ks];
    a.y = h1[m * 16 + 4 * ch + ks + 1];
    b.x = (lo < 8) ? W2s[lo * 16 + 4 * ch + ks] : 0.0f;  // W2s is [8,16]
    b.y = (lo < 8) ? W2s[lo * 16 + 4 * ch + ks + 1] : 0.0f;
    c = __builtin_amdgcn_wmma_f32_16x16x4_f32(false, a, false, b, (short)0, c,
                                              false, false);
  }

  // stage + coalesced h2 store
  __builtin_amdgcn_wave_barrier();
  if (lo < 8) {
#pragma unroll
    for (int j = 0; j < 8; ++j) ldsA[w][j + 8 * hi][lo] = c[j];
  }
  __builtin_amdgcn_wave_barrier();
  if (full) {
    int idx = lane * 4;
    int r = idx >> 3, cc = idx & 7;
    float4 v = make_float4(ldsA[w][r][cc], ldsA[w][r][cc + 1],
                           ldsA[w][r][cc + 2], ldsA[w][r][cc + 3]);
    *reinterpret_cast<float4*>(&h2[base * 8 + idx]) = v;
  } else if (lo < 8) {
#pragma unroll
    for (int j = 0; j < 8; ++j) {
      int node = base + j + 8 * hi;
      if (node < N) h2[node * 8 + lo] = c[j];
    }
  }
}

// ---------------------------------------------------------------------------
__global__ __launch_bounds__(TPB) void k_score(const float4* __restrict__ h2,
                                               const int* __restrict__ src,
                                               const int* __restrict__ dst,
                                               float* __restrict__ out, int E) {
  int e = blockIdx.x * TPB + threadIdx.x;
  if (e >= E) return;
  int s = src[e], d = dst[e];
  float4 s0 = h2[s * 2 + 0], s1 = h2[s * 2 + 1];
  float4 d0 = h2[d * 2 + 0], d1 = h2[d * 2 + 1];
  float acc = s0.x * d0.x + s0.y * d0.y + s0.z * d0.z + s0.w * d0.w +
              s1.x * d1.x + s1.y * d1.y + s1.z * d1.z + s1.w * d1.w;
  __builtin_nontemporal_store(acc, &out[e]);  // 64 MB stream-out, skip caches
}

// ===========================================================================
extern "C" void kernel_launch(void* const* d_in, const int* in_sizes, int n_in,
                              void* d_out, int out_size, void* d_ws,
                              size_t ws_size, hipStream_t stream) {
  const float* x = (const float*)d_in[0];
  const int* src = (const int*)d_in[1];
  const int* dst = (const int*)d_in[2];
  const float* W1s = (const float*)d_in[3];
  const float* W1n = (const float*)d_in[4];
  const float* b1 = (const float*)d_in[5];
  const float* W2s = (const float*)d_in[6];
  const float* W2n = (const float*)d_in[7];
  const float* b2 = (const float*)d_in[8];
  float* out = (float*)d_out;

  const int N = in_sizes[0] / 4;
  const int E = in_sizes[1];

  float* ws = (float*)d_ws;
  float* deg = ws;                      // N
  float* acc1 = deg + (size_t)N;        // 4N
  float* acc2 = acc1 + (size_t)4 * N;   // 8N
  float* h1 = acc2 + (size_t)8 * N;     // 16N
  float* p2 = h1 + (size_t)16 * N;      // 8N
  float* h2 = p2 + (size_t)8 * N;       // 8N
  if (ws_size < (size_t)45 * N * sizeof(float)) return;

  // 1) zero accumulators (deg | acc1 | acc2 contiguous: 13N floats)
  const int nz = 13 * N;
  const int n4 = nz / 4;
  k_zero<<<2048, TPB, 0, stream>>>((float4*)deg, n4, deg + (size_t)n4 * 4,
                                   nz - n4 * 4);

  const int eBlocks = (E + TPB - 1) / TPB;
  const int numTiles = (N + 15) / 16;
  const int nBlocks = (numTiles + WPB - 1) / WPB;

  // 2) layer-1 edge aggregation
  k_edge1<<<eBlocks, TPB, 0, stream>>>((const float4*)x, src, dst, acc1, deg, E);
  // 3) layer-1 node update (+ fused layer-2 neighbor projection)
  k_node1<<<nBlocks, TPB, 0, stream>>>(x, acc1, deg, W1s, W1n, b1, W2n, h1, p2,
                                       numTiles, N);
  // 4) layer-2 edge aggregation in projected 8-dim space
  k_edge2<<<eBlocks, TPB, 0, stream>>>((const float4*)p2, src, dst, acc2, E);
  // 5) layer-2 node update
  k_node2<<<nBlocks, TPB, 0, stream>>>(h1, acc2, deg, W2s, b2, h2, numTiles, N);
  // 6) edge scoring
  k_score<<<eBlocks, TPB, 0, stream>>>((const float4*)h2, src, dst, out, E);
}